// BlittingStrokeModel_53102975647761
// MI455X (gfx1250) — compile-verified
//
#include <hip/hip_runtime.h>

// BlittingStrokeModel on gfx1250: AA line rasterization via f32 WMMA.
// Shapes fixed per reference setup_inputs(): B=8, C=3, H=512, W=512, K=17.
//
// Formulation: for pixel features f = [1, col, row, col^2+row^2] (K=4),
//   F2 = wx*dx + wy*dy = f . [-(p0.d), dx, dy, 0]
//   F1 = wx^2 + wy^2   = f . [|p0|^2, -2*p0x, -2*p0y, 1]
// computed as D = A x B with A = (16 segments x 4 coeffs),
// B = (4 features x 16 pixels) via V_WMMA_F32_16X16X4_F32.
// C layout: lane holds ONE pixel (N=lane%16) and 8 segments (M=j+8*half),
// so max-over-segments = 7 lane-local maxes + one shfl_xor(16).

typedef float v2f __attribute__((ext_vector_type(2)));
typedef float v8f __attribute__((ext_vector_type(8)));

#define BB 8
#define CC 3
#define HH 512
#define WW 512
#define KK 17
#define NSEG (KK - 1)   // 16 segments == WMMA M dimension exactly

__global__ __launch_bounds__(256) void stroke_wmma_kernel(
    const float* __restrict__ traj,   // (B, K, 4): [t, x, y, z]
    const int*   __restrict__ lwp,    // scalar line_width
    float*       __restrict__ out)    // (B, C, H, W)
{
    const int row  = blockIdx.x;
    const int b    = blockIdx.y;
    const int tid  = threadIdx.x;
    const int lane = tid & 31;
    const int wave = tid >> 5;
    const int n    = lane & 15;     // this lane's slot in the "16" dim
    const bool hi  = (lane >= 16);  // half selects K-pair (A/B) / M-offset (C)

    const float lwf = (float)(*lwp) + 0.5f;

    // ---- segment parameters for segment index n (lane-owned) ----
    const float* t0 = traj + ((size_t)b * KK + n) * 4;
    const float p0x = t0[1];
    const float p0y = t0[2];
    const float p1x = t0[5];
    const float p1y = t0[6];

    const float dx  = p1x - p0x;
    const float dy  = p1y - p0y;
    const float den = dx * dx + dy * dy + 1e-8f;

    // ---- A matrices: 16 segments (M=lane%16) x 4 coeffs ----
    // 32-bit A 16x4 layout: lanes 0-15 hold K={0,1}, lanes 16-31 hold K={2,3}.
    // F2 coeffs: k0 = -(p0.d), k1 = dx, k2 = dy, k3 = 0
    v2f a2;
    a2.x = hi ? dy   : -(p0x * dx + p0y * dy);
    a2.y = hi ? 0.0f : dx;
    // F1 coeffs: k0 = |p0|^2, k1 = -2*p0x, k2 = -2*p0y, k3 = 1
    v2f a1;
    a1.x = hi ? (-2.0f * p0y) : (p0x * p0x + p0y * p0y);
    a1.y = hi ? 1.0f          : (-2.0f * p0x);

    // ---- per-segment den/invden broadcast for the epilogue ----
    // epilogue segment for VGPR j is m = j + 8*half; lane j+8*half (half 0)
    // owns that segment's den. Hoisted: done once per wave (8 ds ops).
    const int srcBase = hi ? 8 : 0;
    float denv[8], invv[8];
    #pragma unroll
    for (int j = 0; j < 8; ++j) {
        denv[j] = __shfl(den, srcBase + j, 32);
        invv[j] = __builtin_amdgcn_rcpf(denv[j]);   // v_rcp_f32, ~1 ulp
    }

    const float rowf = (float)row;
    const v8f zero = {};
    const size_t hw = (size_t)HH * WW;

    // each wave owns 64 columns of this row: 4 tiles of 16
    #pragma unroll
    for (int t = 0; t < 4; ++t) {
        const int colBase = wave * 64 + t * 16;
        const int col  = colBase + n;            // B: pixel N = lane%16
        const float colf = (float)col;

        // B matrix (K=4 x N=16 pixels): features [1, col, row, col^2+row^2]
        // VGPR g holds row g + 2*(lane/16).
        v2f bf;
        bf.x = hi ? rowf : 1.0f;
        bf.y = hi ? (colf * colf + rowf * rowf) : colf;

        // D = A x B: C[m=segment][n=pixel], 16x16 tiles for F2 and F1
        v8f c2 = __builtin_amdgcn_wmma_f32_16x16x4_f32(
            false, a2, false, bf, (short)0, zero, false, false);
        v8f c1 = __builtin_amdgcn_wmma_f32_16x16x4_f32(
            false, a1, false, bf, (short)0, zero, false, false);

        // epilogue: coverage for this lane's pixel vs 8 segments, lane-local max
        float acc = 0.0f;
        #pragma unroll
        for (int j = 0; j < 8; ++j) {
            const float f2 = c2[j];
            const float f1 = c1[j];
            float tt = fminf(fmaxf(f2 * invv[j], 0.0f), 1.0f);
            // dist^2 = f1 - 2*t*f2 + t^2*den
            float d2 = fmaf(tt, fmaf(tt, denv[j], -2.0f * f2), f1);
            d2 = fmaxf(d2, 0.0f);
            const float dd  = __builtin_amdgcn_sqrtf(d2 + 1e-12f); // v_sqrt_f32
            const float cov = fminf(fmaxf(lwf - dd, 0.0f), 1.0f);
            acc = fmaxf(acc, cov);
        }
        // merge the two halves (segments 0-7 vs 8-15): one cross-lane op
        const float stroke = fmaxf(acc, __shfl_xor(acc, 16, 32));

        // coalesced stores: lanes 0-15 -> ch0 (16 contiguous cols),
        // lanes 16-31 -> ch1, lanes 0-15 again -> ch2.
        const size_t pix = (size_t)row * WW + col;
        const int ch = hi ? 1 : 0;
        out[((size_t)(b * CC + ch)) * hw + pix] = stroke;
        if (!hi)
            out[((size_t)(b * CC + 2)) * hw + pix] = stroke;
    }
}

extern "C" void kernel_launch(void* const* d_in, const int* in_sizes, int n_in,
                              void* d_out, int out_size, void* d_ws, size_t ws_size,
                              hipStream_t stream) {
    (void)in_sizes; (void)n_in; (void)d_ws; (void)ws_size; (void)out_size;
    // d_in[0] = images (unused: output depends only on shape)
    const float* traj = (const float*)d_in[1];
    const int*   lw   = (const int*)d_in[2];
    float*       out  = (float*)d_out;

    dim3 grid(HH, BB);   // one block per (row, batch); 8 waves cover 512 cols
    stroke_wmma_kernel<<<grid, 256, 0, stream>>>(traj, lw, out);
}